// YOLOLoss_29583734735163
// MI455X (gfx1250) — compile-verified
//
#include <hip/hip_runtime.h>
#include <math.h>

#define BATCH   16
#define NCLS    80
#define NTGT    128
#define NTOTAL  403200   // total cells (48*6400 + 48*1600 + 48*400)
#define NV4     100800   // NTOTAL / 4 (float4 elements)
#define NV4_S0  76800    // 48 * 1600  (pred0 float4s)
#define NV4_S01 96000    // + 48 * 400 (pred1 float4s)
#define NB2     256      // blocks for bulk noobj kernel

typedef float v2f __attribute__((ext_vector_type(2)));
typedef float v8f __attribute__((ext_vector_type(8)));

// ---------------------------------------------------------------------------
// f32 wave reduction via V_WMMA_F32_16X16X4_F32 (wave32, full precision).
// A[m][0] = p[m] (lanes 0-15), A[m][2] = p[m+16] (lanes 16-31), B = ones.
// D[m][n] = p_m + p_{m+16}; lane L holds C vgprs r=0..7 -> rows r (L<16) or
// r+8 (L>=16). Sum of the 8 regs + shfl_xor(16) = full 32-lane sum in every
// lane. Requires EXEC all-ones (call sites guarantee it).
// ---------------------------------------------------------------------------
__device__ __forceinline__ float wave_sum_wmma(float p) {
    v2f a; a[0] = p;    a[1] = 0.0f;
    v2f b; b[0] = 1.0f; b[1] = 1.0f;
    v8f c = {0.f, 0.f, 0.f, 0.f, 0.f, 0.f, 0.f, 0.f};
    c = __builtin_amdgcn_wmma_f32_16x16x4_f32(false, a, false, b, (short)0, c,
                                              false, false);
    float s = c[0] + c[1] + c[2] + c[3] + c[4] + c[5] + c[6] + c[7];
    s += __shfl_xor(s, 16, 32);
    return s;
}

__device__ __forceinline__ float softplus_core(float x) {
    // log1p(exp(-|x|))
    return log1pf(expf(-fabsf(x)));
}
__device__ __forceinline__ float bce0(float x) {  // BCE(x, z=0)
    return fmaxf(x, 0.0f) + softplus_core(x);
}

// ---------------------------------------------------------------------------
// Kernel 1: target assignment (1 block, 128 threads; one thread per target).
// ---------------------------------------------------------------------------
__global__ void __launch_bounds__(NTGT)
assign_kernel(const float* __restrict__ targets, const float* __restrict__ anchors,
              int* __restrict__ keepArr, int* __restrict__ cellB, int* __restrict__ cellA,
              int* __restrict__ cellJ, int* __restrict__ cellI, int* __restrict__ cellCls,
              float* __restrict__ tvArr, float* __restrict__ ntOut)
{
    __shared__ int s_lin[3][NTGT];
    __shared__ int s_valid[NTGT];

    const int t = threadIdx.x;
    const float* tr = targets + t * 6;
    const float tb = tr[0], tc = tr[1], x = tr[2], y = tr[3], w = tr[4], h = tr[5];
    const float ssum = tb + tc + x + y + w + h;
    const int b   = (int)tb;
    const int cls = (int)tc;
    const bool valid = (ssum != 0.0f) && (b < BATCH) && (cls < NCLS) &&
                       (x >= 0.f) && (x <= 1.f) && (y >= 0.f) && (y <= 1.f) &&
                       (w > 0.f) && (w <= 1.f) && (h > 0.f) && (h <= 1.f);
    s_valid[t] = valid ? 1 : 0;

    const int Gs[3] = {80, 40, 20};
#pragma unroll
    for (int s = 0; s < 3; ++s) {
        const int   G      = Gs[s];
        const float stride = 640.0f / (float)G;
        const float gx = x * G, gy = y * G, gw = w * G, gh = h * G;
        const int gi = min((int)fminf(fmaxf(gx, 0.f), (float)G), G - 1);
        const int gj = min((int)fminf(fmaxf(gy, 0.f), (float)G), G - 1);

        int best = 0; float bestR = -1.0f, awb = 0.f, ahb = 0.f;
#pragma unroll
        for (int a2 = 0; a2 < 3; ++a2) {
            const float aw = anchors[s * 6 + a2 * 2 + 0] / stride;
            const float ah = anchors[s * 6 + a2 * 2 + 1] / stride;
            const float inter = fminf(gw, aw) * fminf(gh, ah);
            const float uni   = gw * gh + aw * ah - inter + 1e-7f;
            const float r     = inter / uni;
            if (r > bestR) { bestR = r; best = a2; awb = aw; ahb = ah; }
        }
        s_lin[s][t] = ((b * 3 + best) * G + gj) * G + gi;

        const int o = s * NTGT + t;
        cellB[o] = b; cellA[o] = best; cellJ[o] = gj; cellI[o] = gi; cellCls[o] = cls;
        tvArr[o * 4 + 0] = gx - (float)gi;
        tvArr[o * 4 + 1] = gy - (float)gj;
        tvArr[o * 4 + 2] = logf(gw / (awb + 1e-16f) + 1e-16f);
        tvArr[o * 4 + 3] = logf(gh / (ahb + 1e-16f) + 1e-16f);
    }
    __syncthreads();

#pragma unroll
    for (int s = 0; s < 3; ++s) {
        const int lin = s_lin[s][t];
        bool over = false;
        for (int j = t + 1; j < NTGT; ++j)
            over = over || ((s_lin[s][j] == lin) && s_valid[j]);
        keepArr[s * NTGT + t] = (s_valid[t] && !over) ? 1 : 0;
    }
    if (t == 0) {
        int V = 0;
        for (int j = 0; j < NTGT; ++j) V += s_valid[j];
        ntOut[0] = fmaxf((float)(3 * V), 1.0f);
    }
}

// ---------------------------------------------------------------------------
// Kernel 2: bulk noobj reduction over conf channels only (1.6 MB total),
// vectorized as float4 (global_load_b128): every conf-plane offset
// (b*255+a*85+4)*G^2 is a multiple of 4 floats -> 16B-aligned.
// Deterministic: per-block partials, no atomics. WMMA wave reduce.
// ---------------------------------------------------------------------------
__global__ void __launch_bounds__(256)
noobj_kernel(const float* __restrict__ p0, const float* __restrict__ p1,
             const float* __restrict__ p2, float* __restrict__ partials)
{
    const int tid = blockIdx.x * blockDim.x + threadIdx.x;
    const int nth = gridDim.x * blockDim.x;

    float acc = 0.0f;
    for (int idx = tid; idx < NV4; idx += nth) {
        float4 v;
        if (idx < NV4_S0) {
            const int i2 = idx;
            const int plane = i2 / 1600;          // const-divisor -> mul/shift
            const int rem   = i2 - plane * 1600;  // float4 index in 80x80 plane
            const int b = plane / 3, a = plane - 3 * b;
            const float4* bp =
                reinterpret_cast<const float4*>(p0 + (b * 255 + a * 85 + 4) * 6400);
            v = bp[rem];
        } else if (idx < NV4_S01) {
            const int i2 = idx - NV4_S0;
            const int plane = i2 / 400;
            const int rem   = i2 - plane * 400;
            const int b = plane / 3, a = plane - 3 * b;
            const float4* bp =
                reinterpret_cast<const float4*>(p1 + (b * 255 + a * 85 + 4) * 1600);
            v = bp[rem];
        } else {
            const int i2 = idx - NV4_S01;
            const int plane = i2 / 100;
            const int rem   = i2 - plane * 100;
            const int b = plane / 3, a = plane - 3 * b;
            const float4* bp =
                reinterpret_cast<const float4*>(p2 + (b * 255 + a * 85 + 4) * 400);
            v = bp[rem];
        }
        acc += bce0(v.x) + bce0(v.y) + bce0(v.z) + bce0(v.w);
    }

    const float wsum = wave_sum_wmma(acc);   // all lanes active here
    __shared__ float sw[8];
    const int lane = threadIdx.x & 31, wv = threadIdx.x >> 5;
    if (lane == 0) sw[wv] = wsum;
    __syncthreads();
    if (threadIdx.x == 0) {
        float s = 0.f;
        for (int i = 0; i < 8; ++i) s += sw[i];
        partials[blockIdx.x] = s;
    }
}

// ---------------------------------------------------------------------------
// Kernel 3: per-kept-target losses. 384 blocks x 128 threads.
//   tid<80: class BCE; tid 80..83: box term; tid 84: lobj correction.
// ---------------------------------------------------------------------------
__global__ void __launch_bounds__(128)
target_kernel(const float* __restrict__ p0, const float* __restrict__ p1,
              const float* __restrict__ p2,
              const int* __restrict__ keepArr, const int* __restrict__ cellB,
              const int* __restrict__ cellA, const int* __restrict__ cellJ,
              const int* __restrict__ cellI, const int* __restrict__ cellCls,
              const float* __restrict__ tvArr,
              float* __restrict__ tpBox, float* __restrict__ tpObj,
              float* __restrict__ tpCls)
{
    const int k   = blockIdx.x;   // 0..383  (scale = k>>7)
    const int tid = threadIdx.x;  // 0..127
    __shared__ float sm[3 * 128];

    float cterm = 0.f, bterm = 0.f, oterm = 0.f;
    if (keepArr[k]) {
        const int s = k >> 7;
        const float* base = (s == 0) ? p0 : (s == 1) ? p1 : p2;
        const int G  = (s == 0) ? 80 : (s == 1) ? 40 : 20;
        const int G2 = G * G;
        const int b = cellB[k], a = cellA[k], gj = cellJ[k], gi = cellI[k];
        const int cls = cellCls[k];
        const int cellOff  = gj * G + gi;
        const int chanBase = b * 255 + a * 85;

        if (tid < 80) {
            const float pv = base[(chanBase + 5 + tid) * G2 + cellOff];
            const float z  = (tid == cls) ? 1.0f : 0.0f;
            cterm = fmaxf(pv, 0.f) - pv * z + softplus_core(pv);
        } else if (tid < 84) {
            const int kk = tid - 80;
            const float pv = base[(chanBase + kk) * G2 + cellOff];
            const float tv = tvArr[k * 4 + kk];
            const float v  = (kk < 2) ? (1.0f / (1.0f + expf(-pv))) : pv;
            const float d  = v - tv;
            bterm = d * d;
        } else if (tid == 84) {
            const float pv = base[(chanBase + 4) * G2 + cellOff];
            const float sp = softplus_core(pv);
            const float b1 = fmaxf(pv, 0.f) - pv + sp;   // BCE(conf, 1)
            const float b0 = fmaxf(pv, 0.f) + sp;        // BCE(conf, 0)
            oterm = b1 - 0.5f * b0;                      // replaces noobj term
        }
    }
    sm[tid] = cterm; sm[128 + tid] = bterm; sm[256 + tid] = oterm;
    __syncthreads();
    for (int st = 64; st > 0; st >>= 1) {
        if (tid < st) {
            sm[tid]       += sm[tid + st];
            sm[128 + tid] += sm[128 + tid + st];
            sm[256 + tid] += sm[256 + tid + st];
        }
        __syncthreads();
    }
    if (tid == 0) { tpCls[k] = sm[0]; tpBox[k] = sm[128]; tpObj[k] = sm[256]; }
}

// ---------------------------------------------------------------------------
// Kernel 4: deterministic final combine -> d_out[0].
// ---------------------------------------------------------------------------
__global__ void __launch_bounds__(256)
final_kernel(const float* __restrict__ noobjPart, const float* __restrict__ tpBox,
             const float* __restrict__ tpObj, const float* __restrict__ tpCls,
             const float* __restrict__ ntF, float* __restrict__ out)
{
    __shared__ float s0[256], s1[256], s2[256], s3[256];
    const int tid = threadIdx.x;

    float aN = noobjPart[tid];   // NB2 == 256
    float aB = 0.f, aO = 0.f, aC = 0.f;
    for (int i = tid; i < 384; i += 256) {
        aB += tpBox[i]; aO += tpObj[i]; aC += tpCls[i];
    }
    s0[tid] = aN; s1[tid] = aB; s2[tid] = aO; s3[tid] = aC;
    __syncthreads();
    for (int st = 128; st > 0; st >>= 1) {
        if (tid < st) {
            s0[tid] += s0[tid + st]; s1[tid] += s1[tid + st];
            s2[tid] += s2[tid + st]; s3[tid] += s3[tid + st];
        }
        __syncthreads();
    }
    if (tid == 0) {
        const float nt   = ntF[0];
        const float lobj = 0.5f * s0[0] + s2[0];
        out[0] = s1[0] / nt + lobj / (float)NTOTAL + s3[0] / nt;
    }
}

// ---------------------------------------------------------------------------
// Host launcher. Inputs: pred0, pred1, pred2, targets, anchors (all f32).
// ---------------------------------------------------------------------------
extern "C" void kernel_launch(void* const* d_in, const int* in_sizes, int n_in,
                              void* d_out, int out_size, void* d_ws, size_t ws_size,
                              hipStream_t stream) {
    const float* p0 = (const float*)d_in[0];
    const float* p1 = (const float*)d_in[1];
    const float* p2 = (const float*)d_in[2];
    const float* targets = (const float*)d_in[3];
    const float* anchors = (const float*)d_in[4];
    float* out = (float*)d_out;

    // Workspace layout (floats then ints then floats; ~21 KB total).
    float* ws_f      = (float*)d_ws;
    float* noobjPart = ws_f;               // 256
    float* tpBox     = ws_f + 256;         // 384
    float* tpObj     = ws_f + 640;         // 384
    float* tpCls     = ws_f + 1024;        // 384
    float* ntF       = ws_f + 1408;        // 1
    int*   ws_i      = (int*)(ws_f + 1412);
    int*   keepArr   = ws_i;               // 384
    int*   cellB     = ws_i + 384;
    int*   cellA     = ws_i + 768;
    int*   cellJ     = ws_i + 1152;
    int*   cellI     = ws_i + 1536;
    int*   cellCls   = ws_i + 1920;
    float* tvArr     = (float*)(ws_i + 2304);  // 384*4

    assign_kernel<<<1, NTGT, 0, stream>>>(targets, anchors, keepArr, cellB, cellA,
                                          cellJ, cellI, cellCls, tvArr, ntF);
    noobj_kernel<<<NB2, 256, 0, stream>>>(p0, p1, p2, noobjPart);
    target_kernel<<<384, 128, 0, stream>>>(p0, p1, p2, keepArr, cellB, cellA,
                                           cellJ, cellI, cellCls, tvArr,
                                           tpBox, tpObj, tpCls);
    final_kernel<<<1, 256, 0, stream>>>(noobjPart, tpBox, tpObj, tpCls, ntF, out);
}